// AnatomicalTextEnhancer_76124000354852
// MI455X (gfx1250) — compile-verified
//
#include <hip/hip_runtime.h>
#include <hip/hip_bf16.h>
#include <math.h>

// ---------------------------------------------------------------------------
// Cosine-sim retrieval + top-k for MI455X (gfx1250, wave32).
//   query      [B=16, R=29, H=768]  fp32
//   db         [R=29, N=8192, H=768] fp32
//   exclude    [B, R] int32
// Outputs (concatenated fp32): top_vals [B,R,8], top_idx [B,R,8], best [B,R]
//
// Memory-bound: 730MB db stream -> ~31us floor @ 23.3TB/s. Compute uses
// V_WMMA_F32_16X16X4_F32 (B=16 == WMMA M). K-steps are preloaded in blocks
// of 8 (16 b64 loads in flight per wave) so Little's law is satisfied with
// 928 workgroups; db rows' L2 norms accumulate in-flight and query norms
// fold into the epilogue scale.
// ---------------------------------------------------------------------------

typedef __attribute__((ext_vector_type(2))) float v2f;
typedef __attribute__((ext_vector_type(8))) float v8f;

#define NEG_INF_V (-1e30f)

constexpr int Bb    = 16;
constexpr int Rr    = 29;
constexpr int Nn    = 8192;
constexpr int Hh    = 768;
constexpr int TOPK  = 8;
constexpr int CHUNK = 256;            // columns per workgroup
constexpr int NCH   = Nn / CHUNK;     // 32 chunks per region
constexpr int THREADS = 256;          // 8 waves (wave32)
constexpr int WAVES   = THREADS / 32;
constexpr int TPW     = CHUNK / (WAVES * 16); // tiles per wave = 2
constexpr int SSTRIDE = CHUNK + 4;    // LDS row pad -> conflict-free halves
constexpr int KBLK    = 8;            // K-steps preloaded per block (h += 32)

__device__ __forceinline__ void topk_insert(float v, int id,
                                            float tv[TOPK], int ti[TOPK]) {
  if (v <= tv[TOPK - 1]) return;
  float cv = v; int ci = id;
#pragma unroll
  for (int p = 0; p < TOPK; ++p) {
    if (cv > tv[p]) {
      float tf = tv[p]; int tn = ti[p];
      tv[p] = cv; ti[p] = ci;
      cv = tf; ci = tn;
    }
  }
}

__global__ __launch_bounds__(THREADS)
void sim_topk_stageA(const float* __restrict__ q, const float* __restrict__ db,
                     const int* __restrict__ ex,
                     float* __restrict__ pv, int* __restrict__ pi) {
  __shared__ float sims[Bb * SSTRIDE];   // ~17KB, reused as candidate buffer
  __shared__ float red[THREADS];
  __shared__ float qinv[Bb];
  __shared__ int   exl[Bb];

  const int r     = blockIdx.x / NCH;
  const int chunk = blockIdx.x % NCH;
  const int t     = threadIdx.x;

  // ---- query inverse norms for this region (q stays raw; scale in epilogue)
  {
    const int b = t >> 4, hl = t & 15;
    const float* qb = q + ((size_t)b * Rr + r) * Hh;
    float s = 0.f;
#pragma unroll 4
    for (int i = 0; i < Hh / 16; ++i) { float x = qb[hl + 16 * i]; s += x * x; }
    red[t] = s;
  }
  if (t < Bb) exl[t] = ex[t * Rr + r];
  __syncthreads();
  if (t < Bb) {
    float s = 0.f;
#pragma unroll
    for (int i = 0; i < 16; ++i) s += red[t * 16 + i];
    qinv[t] = 1.0f / fmaxf(sqrtf(s), 1e-12f);
  }
  __syncthreads();

  // ---- WMMA f32 16x16x4 tiles: M=batch(16), N=16 db rows, K over H=768
  const int wave  = t >> 5;
  const int lane  = t & 31;
  const int lrow  = lane & 15;          // M (for A) / N (for B) within tile
  const int khalf = (lane >> 4) << 1;   // K pair base: 0 or 2
  const float* qrow = q + ((size_t)lrow * Rr + r) * Hh + khalf;

  for (int tt = 0; tt < TPW; ++tt) {
    const int col0 = wave * (TPW * 16) + tt * 16;
    const int n0   = chunk * CHUNK + col0;
    const float* drow = db + ((size_t)r * Nn + n0 + lrow) * Hh + khalf;

    v8f  c  = {};
    float ss = 0.f;
    for (int h0 = 0; h0 < Hh; h0 += 4 * KBLK) {
      // issue all loads for this K-block first -> 2*KBLK b64 loads in flight
      v2f a[KBLK], bd[KBLK];
#pragma unroll
      for (int u = 0; u < KBLK; ++u) {
        a[u]  = *(const v2f*)(qrow + h0 + 4 * u);
        bd[u] = __builtin_nontemporal_load((const v2f*)(drow + h0 + 4 * u));
      }
      // then consume: WMMA chain + db row sum-of-squares
#pragma unroll
      for (int u = 0; u < KBLK; ++u) {
        ss = fmaf(bd[u].x, bd[u].x, ss);
        ss = fmaf(bd[u].y, bd[u].y, ss);
        // (neg_a, A, neg_b, B, c_mod, C, reuse_a, reuse_b)
        c = __builtin_amdgcn_wmma_f32_16x16x4_f32(false, a[u], false, bd[u],
                                                  (short)0, c, false, false);
      }
    }
    // combine K-half sums-of-squares: every lane gets its column's full norm
    ss += __shfl_xor(ss, 16, 32);
    const float dinv  = 1.0f / fmaxf(sqrtf(ss), 1e-12f);
    const int   mbase = (lane >> 4) << 3; // C layout: VGPR j -> M=j (lo), j+8 (hi)
#pragma unroll
    for (int j = 0; j < 8; ++j) {
      const int m = mbase + j;
      float v = c[j] * dinv * qinv[m];
      if (n0 + lrow == exl[m]) v = NEG_INF_V;
      sims[m * SSTRIDE + col0 + lrow] = v;
    }
  }
  __syncthreads();

  // ---- per-(batch, 16-col segment) top-8 in registers
  float tv[TOPK]; int ti[TOPK];
#pragma unroll
  for (int k = 0; k < TOPK; ++k) { tv[k] = -3.4e38f; ti[k] = 0; }
  {
    const int b = t >> 4, seg = t & 15;
    const int seglen = CHUNK / 16;
    const float* srow = sims + b * SSTRIDE + seg * seglen;
    const int ibase = chunk * CHUNK + seg * seglen;
    for (int i = 0; i < seglen; ++i) topk_insert(srow[i], ibase + i, tv, ti);
  }
  __syncthreads();
  // reuse sims LDS as candidate buffer: 2048 vals + 2048 idx (fits in 16*260)
  float* candV = sims;
  int*   candI = (int*)(sims + Bb * 16 * TOPK);
  {
    const int b = t >> 4, seg = t & 15;
#pragma unroll
    for (int k = 0; k < TOPK; ++k) {
      candV[(b * 16 + seg) * TOPK + k] = tv[k];
      candI[(b * 16 + seg) * TOPK + k] = ti[k];
    }
  }
  __syncthreads();

  // ---- one thread per batch merges 16 sorted lists -> partial top-8
  if (t < Bb) {
    const int b = t;
    float mv[TOPK]; int mi[TOPK];
#pragma unroll
    for (int k = 0; k < TOPK; ++k) { mv[k] = -3.4e38f; mi[k] = 0; }
    for (int s = 0; s < 16; ++s) {
      for (int k = 0; k < TOPK; ++k) {
        float v = candV[(b * 16 + s) * TOPK + k];
        if (v <= mv[TOPK - 1]) break;           // lists are sorted desc
        topk_insert(v, candI[(b * 16 + s) * TOPK + k], mv, mi);
      }
    }
    float* opv = pv + ((size_t)(r * NCH + chunk) * Bb + b) * TOPK;
    int*   opi = pi + ((size_t)(r * NCH + chunk) * Bb + b) * TOPK;
#pragma unroll
    for (int k = 0; k < TOPK; ++k) { opv[k] = mv[k]; opi[k] = mi[k]; }
  }
}

__global__ __launch_bounds__(256)
void sim_topk_stageB(const float* __restrict__ pv, const int* __restrict__ pi,
                     float* __restrict__ out) {
  const int i = blockIdx.x * blockDim.x + threadIdx.x;
  if (i >= Bb * Rr) return;
  const int b = i / Rr, r = i % Rr;

  float tv[TOPK]; int ti[TOPK];
#pragma unroll
  for (int k = 0; k < TOPK; ++k) { tv[k] = -3.4e38f; ti[k] = 0; }

  for (int ch = 0; ch < NCH; ++ch) {
    const float* cv = pv + ((size_t)(r * NCH + ch) * Bb + b) * TOPK;
    const int*   ci = pi + ((size_t)(r * NCH + ch) * Bb + b) * TOPK;
    for (int k = 0; k < TOPK; ++k) {
      float v = cv[k];
      if (v <= tv[TOPK - 1]) break;             // sorted desc
      topk_insert(v, ci[k], tv, ti);
    }
  }

  float* tvo = out;                              // top_vals [B,R,8]
  float* tio = out + (size_t)Bb * Rr * TOPK;     // top_idx  [B,R,8] (as float)
  float* bso = out + (size_t)2 * Bb * Rr * TOPK; // best     [B,R]
#pragma unroll
  for (int k = 0; k < TOPK; ++k) {
    tvo[(size_t)i * TOPK + k] = tv[k];
    tio[(size_t)i * TOPK + k] = (float)ti[k];
  }
  bso[i] = tv[0];
}

extern "C" void kernel_launch(void* const* d_in, const int* in_sizes, int n_in,
                              void* d_out, int out_size, void* d_ws, size_t ws_size,
                              hipStream_t stream) {
  const float* q  = (const float*)d_in[0];
  const float* db = (const float*)d_in[1];
  const int*   ex = (const int*)d_in[2];
  // d_in[3] = top_k (==8, hardcoded)

  const size_t npart = (size_t)Rr * NCH * Bb * TOPK; // 118784
  float* pv = (float*)d_ws;
  int*   pi = (int*)((char*)d_ws + npart * sizeof(float));

  sim_topk_stageA<<<dim3(Rr * NCH), dim3(THREADS), 0, stream>>>(q, db, ex, pv, pi);
  sim_topk_stageB<<<dim3((Bb * Rr + 255) / 256), dim3(256), 0, stream>>>(
      pv, pi, (float*)d_out);
}